// SpectrumMemoryBank_47193100648812
// MI455X (gfx1250) — compile-verified
//
#include <hip/hip_runtime.h>

#define MEMORY_DIM   256
#define NUM_BANKS    16
#define SPECTRAL_DIM 8
#define LN_EPS       1e-5f
#define BM           32                       // batch rows per workgroup
#define CHUNK        32                       // K-rows per TDM chunk
#define NCHUNK_BANK  (MEMORY_DIM / CHUNK)     // 8
#define NCHUNK_TOT   (NUM_BANKS * NCHUNK_BANK) // 128

typedef float v2f __attribute__((ext_vector_type(2)));
typedef float v8f __attribute__((ext_vector_type(8)));
typedef unsigned int u32x4 __attribute__((ext_vector_type(4)));
typedef int i32x4 __attribute__((ext_vector_type(4)));
typedef int i32x8 __attribute__((ext_vector_type(8)));

#if defined(__has_builtin)
# if __has_builtin(__builtin_amdgcn_tensor_load_to_lds)
#  define USE_TDM 1
# endif
#endif
#ifndef USE_TDM
# define USE_TDM 0
#endif

// D = A(16x4) * B(4x16) + C, fp32, wave32.
#define WMMA_F32(A, B, C) \
    __builtin_amdgcn_wmma_f32_16x16x4_f32(false, (A), false, (B), (short)0, (C), false, false)

__device__ __forceinline__ float fast_tanh(float x) {
    // tanh(x) = 1 - 2/(exp(2x)+1)   (v_exp_f32 + v_rcp_f32)
    const float e = __expf(2.0f * x);
    return 1.0f - 2.0f * __builtin_amdgcn_rcpf(e + 1.0f);
}

#if USE_TDM
// Issue one TDM 2D tile copy: CHUNK rows x 256 cols of f32, row-major,
// global -> LDS. Descriptor per CDNA5 ISA section 8 (D# groups 0/1).
// This toolchain's builtin takes 6 args: (g0, g1, g2, g3, g4, cpol).
__device__ __forceinline__ void tdm_load_chunk(const float* gsrc, unsigned ldsByteOff) {
    const unsigned long long ga = (unsigned long long)(uintptr_t)gsrc;
    u32x4 g0;
    g0[0] = 1u;                                  // count=1, is_restore=0, gather off
    g0[1] = ldsByteOff;                          // lds_addr (bytes)
    g0[2] = (unsigned)ga;                        // global_addr[31:0]
    g0[3] = (unsigned)(ga >> 32) | (2u << 30);   // global_addr[56:32] | type=2
    i32x8 g1;
    g1[0] = (2 << 16);                           // data_size = 4 bytes; no multicast
    g1[1] = (int)((MEMORY_DIM & 0xFFFF) << 16);  // tensor_dim0[15:0]  (=256)
    g1[2] = (int)((MEMORY_DIM & 0xFFFF) << 16);  // dim0[31:16]=0 | tensor_dim1[15:0]
    g1[3] = (int)(((unsigned)MEMORY_DIM) << 16); // dim1[31:16]=0 | tile_dim0 = 256
    g1[4] = CHUNK;                               // tile_dim1 = 32, tile_dim2 = 0
    g1[5] = MEMORY_DIM;                          // tensor_dim0_stride[31:0] = 256
    g1[6] = 0;                                   // dim0_stride[47:32] | dim1_stride lo
    g1[7] = 0;
    i32x4 g2 = {0, 0, 0, 0};
    i32x4 g3 = {0, 0, 0, 0};
    i32x8 g4 = {0, 0, 0, 0, 0, 0, 0, 0};
    __builtin_amdgcn_tensor_load_to_lds(g0, g1, g2, g3, g4, 0);
}
#endif

__global__ __launch_bounds__(256)
void smb_fused_kernel(const float* __restrict__ spec,
                      const float* __restrict__ query,
                      const float* __restrict__ Wa,
                      const float* __restrict__ ba,
                      const float* __restrict__ Wb,
                      const float* __restrict__ bb,
                      const float* __restrict__ gamma,
                      const float* __restrict__ beta,
                      const float* __restrict__ Wr,
                      const float* __restrict__ br,
                      float* __restrict__ out)
{
    __shared__ float bufs[2 * CHUNK * MEMORY_DIM]; // 64 KB: double-buffered Wb chunks
    __shared__ float qLds[BM * MEMORY_DIM];        // 32 KB: Q tile, reused for combined
    __shared__ float wgtLds[BM * NUM_BANKS];       // 2 KB : logits -> softmax weights
    __shared__ float redSum[BM];
    __shared__ float redSq[BM];

    const int t       = threadIdx.x;
    const int lane    = t & 31;
    const int wave    = t >> 5;
    const int m       = lane & 15;        // row (A/C) or col (B/C) within 16
    const int half    = lane >> 4;        // 0: K=0,1 / M=j ; 1: K=2,3 / M=j+8
    const int rowblk  = (wave & 1) * 16;  // 16-row block within tile
    const int colbase = (wave >> 1) * 64; // 64-col strip
    const long B0     = (long)blockIdx.x * BM;

    // ---------------- Phase 0: stage Q tile into LDS (coalesced b128) --------
    {
        const float4* src = (const float4*)(query + B0 * MEMORY_DIM);
        float4* dst = (float4*)qLds;
        #pragma unroll
        for (int i = 0; i < (BM * MEMORY_DIM / 4) / 256; ++i)
            dst[t + i * 256] = src[t + i * 256];
    }

    // ---------------- Phase 0b: routing logits + softmax ---------------------
    for (int p = t; p < BM * NUM_BANKS; p += 256) {
        const int r = p >> 4, k = p & 15;
        float acc = ba[k];
        const float* sv = spec + (B0 + r) * SPECTRAL_DIM;
        #pragma unroll
        for (int d = 0; d < SPECTRAL_DIM; ++d)
            acc += sv[d] * Wa[d * NUM_BANKS + k];
        wgtLds[p] = acc;
    }
    __syncthreads();
    if (t < BM) {
        float mx = -1e30f;
        #pragma unroll
        for (int k = 0; k < NUM_BANKS; ++k) mx = fmaxf(mx, wgtLds[t * 16 + k]);
        float e[NUM_BANKS], s = 0.f;
        #pragma unroll
        for (int k = 0; k < NUM_BANKS; ++k) { e[k] = __expf(wgtLds[t * 16 + k] - mx); s += e[k]; }
        const float inv = 1.f / s;
        #pragma unroll
        for (int k = 0; k < NUM_BANKS; ++k) wgtLds[t * 16 + k] = e[k] * inv;
    }
    __syncthreads();

    // ---------------- Phase 1: bank loop, TDM-fed GEMM + LN + tanh + combine -
    v8f comb0 = {}, comb1 = {}, comb2 = {}, comb3 = {};

    const float* aPtr = qLds + (rowblk + m) * MEMORY_DIM + half * 2;

#if USE_TDM
    const unsigned bufOff = (unsigned)(uintptr_t)&bufs[0];
    // prologue: DMA chunk 0 into buffer 0, wait, publish
    if (wave == 0) {
        tdm_load_chunk(Wb, bufOff);
        __builtin_amdgcn_s_wait_tensorcnt(0);
    }
    __syncthreads();
#endif

    #pragma unroll 1
    for (int k = 0; k < NUM_BANKS; ++k) {
        if (t < BM) { redSum[t] = 0.f; redSq[t] = 0.f; }

        v8f acc0 = {}, acc1 = {}, acc2 = {}, acc3 = {};

        #pragma unroll 1
        for (int cc = 0; cc < NCHUNK_BANK; ++cc) {
            const int c = k * NCHUNK_BANK + cc;

#if USE_TDM
            // kick off DMA of the next chunk into the other buffer
            if (wave == 0 && c + 1 < NCHUNK_TOT) {
                const int cn = c + 1;
                const float* gsrc = Wb + ((size_t)(cn >> 3) << 16)
                                       + (size_t)(cn & 7) * CHUNK * MEMORY_DIM;
                tdm_load_chunk(gsrc, bufOff + (unsigned)((cn & 1) * CHUNK * MEMORY_DIM * 4));
            }
#else
            // fallback: cooperative copy of chunk c, then publish
            {
                const float4* s4 = (const float4*)(Wb + ((size_t)k << 16)
                                                      + (size_t)cc * CHUNK * MEMORY_DIM);
                float4* d4 = (float4*)(bufs + (c & 1) * (CHUNK * MEMORY_DIM));
                #pragma unroll
                for (int i = 0; i < (CHUNK * MEMORY_DIM / 4) / 256; ++i)
                    d4[t + i * 256] = s4[t + i * 256];
            }
            __syncthreads();
#endif

            // ---- compute on chunk c (B operands from LDS) ----
            const float* bBase = bufs + (c & 1) * (CHUNK * MEMORY_DIM)
                               + (half * 2) * MEMORY_DIM + colbase + m;
            const float* aRun = aPtr + cc * CHUNK;
            #pragma unroll 4
            for (int kl = 0; kl < CHUNK; kl += 4) {
                const v2f a = *(const v2f*)(aRun + kl);
                const float* bp = bBase + kl * MEMORY_DIM;
                v2f b0 = { bp[ 0], bp[ 0 + MEMORY_DIM] };
                v2f b1 = { bp[16], bp[16 + MEMORY_DIM] };
                v2f b2 = { bp[32], bp[32 + MEMORY_DIM] };
                v2f b3 = { bp[48], bp[48 + MEMORY_DIM] };
                acc0 = WMMA_F32(a, b0, acc0);
                acc1 = WMMA_F32(a, b1, acc1);
                acc2 = WMMA_F32(a, b2, acc2);
                acc3 = WMMA_F32(a, b3, acc3);
            }

#if USE_TDM
            if (wave == 0) __builtin_amdgcn_s_wait_tensorcnt(0); // next chunk landed
#endif
            __syncthreads(); // publish next buffer / retire this one
        }

        // + bb (per-column bias; column fixed per accumulator per lane)
        {
            const float* bbk = bb + k * MEMORY_DIM + colbase + m;
            const float c0 = bbk[0], c1 = bbk[16], c2 = bbk[32], c3 = bbk[48];
            #pragma unroll
            for (int j = 0; j < 8; ++j) {
                acc0[j] += c0; acc1[j] += c1; acc2[j] += c2; acc3[j] += c3;
            }
        }

        // per-row sum / sum-of-squares: 16-lane shuffle reduce + LDS float atomics
        #pragma unroll
        for (int j = 0; j < 8; ++j) {
            const float v0 = acc0[j], v1 = acc1[j], v2 = acc2[j], v3 = acc3[j];
            float s1 = v0 + v1 + v2 + v3;
            float s2 = v0 * v0 + v1 * v1 + v2 * v2 + v3 * v3;
            #pragma unroll
            for (int msk = 1; msk < 16; msk <<= 1) {
                s1 += __shfl_xor(s1, msk, 32);
                s2 += __shfl_xor(s2, msk, 32);
            }
            if (m == 0) {
                const int r = rowblk + j + 8 * half;
                atomicAdd(&redSum[r], s1);
                atomicAdd(&redSq[r], s2);
            }
        }
        __syncthreads();

        // LayerNorm (biased var) + fast tanh + softmax-weighted accumulate
        {
            const float* gk = gamma + k * MEMORY_DIM + colbase + m;
            const float* ek = beta  + k * MEMORY_DIM + colbase + m;
            const float g0 = gk[0], g1 = gk[16], g2 = gk[32], g3 = gk[48];
            const float e0 = ek[0], e1 = ek[16], e2 = ek[32], e3 = ek[48];
            #pragma unroll
            for (int j = 0; j < 8; ++j) {
                const int r = rowblk + j + 8 * half;
                const float mu  = redSum[r] * (1.0f / MEMORY_DIM);
                const float var = redSq[r] * (1.0f / MEMORY_DIM) - mu * mu;
                const float rs  = rsqrtf(var + LN_EPS);
                const float w   = wgtLds[r * NUM_BANKS + k];
                comb0[j] += w * fast_tanh((acc0[j] - mu) * rs * g0 + e0);
                comb1[j] += w * fast_tanh((acc1[j] - mu) * rs * g1 + e1);
                comb2[j] += w * fast_tanh((acc2[j] - mu) * rs * g2 + e2);
                comb3[j] += w * fast_tanh((acc3[j] - mu) * rs * g3 + e3);
            }
        }
        __syncthreads();   // red/wgt reads done before next bank re-zeroes
    }

    // ---------------- Phase 2: spill combined into LDS (reuse Q buffer) ------
    #pragma unroll
    for (int j = 0; j < 8; ++j) {
        const int r = rowblk + j + 8 * half;
        qLds[r * MEMORY_DIM + colbase +  0 + m] = comb0[j];
        qLds[r * MEMORY_DIM + colbase + 16 + m] = comb1[j];
        qLds[r * MEMORY_DIM + colbase + 32 + m] = comb2[j];
        qLds[r * MEMORY_DIM + colbase + 48 + m] = comb3[j];
    }
    __syncthreads();

    // ---------------- Phase 3: read head GEMM (combined @ Wr + br) -----------
    {
        v8f o0 = {}, o1 = {}, o2 = {}, o3 = {};
        const float* bPtr = Wr + (half * 2) * MEMORY_DIM + colbase + m;
        #pragma unroll 4
        for (int kk = 0; kk < MEMORY_DIM; kk += 4) {
            const v2f a = *(const v2f*)(aPtr + kk);
            const float* bp = bPtr + kk * MEMORY_DIM;
            v2f b0 = { bp[ 0], bp[ 0 + MEMORY_DIM] };
            v2f b1 = { bp[16], bp[16 + MEMORY_DIM] };
            v2f b2 = { bp[32], bp[32 + MEMORY_DIM] };
            v2f b3 = { bp[48], bp[48 + MEMORY_DIM] };
            o0 = WMMA_F32(a, b0, o0);
            o1 = WMMA_F32(a, b1, o1);
            o2 = WMMA_F32(a, b2, o2);
            o3 = WMMA_F32(a, b3, o3);
        }
        const float* brp = br + colbase + m;
        const float r0 = brp[0], r1 = brp[16], r2 = brp[32], r3 = brp[48];
        #pragma unroll
        for (int j = 0; j < 8; ++j) {
            const long r = B0 + rowblk + j + 8 * half;
            float* op = out + r * MEMORY_DIM + colbase + m;
            op[ 0] = o0[j] + r0;
            op[16] = o1[j] + r1;
            op[32] = o2[j] + r2;
            op[48] = o3[j] + r3;
        }
    }
}

extern "C" void kernel_launch(void* const* d_in, const int* in_sizes, int n_in,
                              void* d_out, int out_size, void* d_ws, size_t ws_size,
                              hipStream_t stream) {
    const float* spec  = (const float*)d_in[0];
    const float* query = (const float*)d_in[1];
    const float* Wa    = (const float*)d_in[2];
    const float* ba    = (const float*)d_in[3];
    const float* Wb    = (const float*)d_in[4];
    const float* bb    = (const float*)d_in[5];
    const float* gamma = (const float*)d_in[6];
    const float* beta  = (const float*)d_in[7];
    const float* Wr    = (const float*)d_in[8];
    const float* br    = (const float*)d_in[9];
    float* out = (float*)d_out;

    const int batch  = in_sizes[1] / MEMORY_DIM;
    const int blocks = batch / BM;

    smb_fused_kernel<<<blocks, 256, 0, stream>>>(spec, query, Wa, ba, Wb, bb,
                                                 gamma, beta, Wr, br, out);
}